// TextEncoder_84911503442642
// MI455X (gfx1250) — compile-verified
//
#include <hip/hip_runtime.h>

#define B_ 32
#define L_ 4
#define W_ 512
#define D_ 768
#define WCHUNKS 4
#define WCHUNK (W_ / WCHUNKS)   // 128 tokens per chunk

typedef __attribute__((ext_vector_type(2))) float v2f;
typedef __attribute__((ext_vector_type(8))) float v8f;

// ---------------------------------------------------------------------------
// Kernel 0: zero the whole output (word slots with no tokens must stay zero,
// and all segment/sent writes are accumulated on top of zeroed memory).
// ---------------------------------------------------------------------------
__global__ void zero_f4(float4* __restrict__ p, int n4) {
  int i = blockIdx.x * blockDim.x + threadIdx.x;
  int stride = gridDim.x * blockDim.x;
  float4 z = {0.f, 0.f, 0.f, 0.f};
  for (; i < n4; i += stride) p[i] = z;
}

// ---------------------------------------------------------------------------
// Kernel 1: fused layer-mean + sorted-segment-sum + transpose-scatter + sent.
//
// Each wave owns a 16-wide d-tile of one batch over one 128-token chunk.
// Per token: one V_WMMA_F32_16X16X4_F32 with
//   A = const 0.25 (16x4)          -> folds the /L layer mean
//   B[k,n] = emb[b, l=k, w, d0+n]  -> lanes 0-15 carry layers {0,1},
//                                     lanes 16-31 carry layers {2,3}
//   C chained across tokens        -> accumulates the segment run sum
// D[m,n] = 0.25 * sum_k B[k,n] is independent of m and of the hardware's
// (VGPR, lane-half)->K mapping, so c[0] in lane n is the finished value for
// column d0+n. Run flushes are 16-lane stores to out[b, d, seg] (stride-W
// scatter, absorbed by L2: the 48 MiB output fits in the 192 MB L2).
// ---------------------------------------------------------------------------
__global__ __launch_bounds__(256) void seg_agg(
    const float* __restrict__ emb,   // [B, L, W, D]
    const int*   __restrict__ seg,   // [B, W], sorted per row
    float*       __restrict__ out_word,  // [B, D, W]
    float*       __restrict__ out_sent)  // [B, D]
{
  const int lane = threadIdx.x & 31;
  const int wave = threadIdx.x >> 5;
  const int b  = blockIdx.z;
  const int ws = blockIdx.y * WCHUNK;
  const int d0 = blockIdx.x * 128 + wave * 16;
  const int n  = lane & 15;
  const int h  = lane >> 4;            // 0 -> layers 0,1 ; 1 -> layers 2,3

  // per-lane pointer at (b, l=2h, w=ws, d0+n); +W_*D_ floats reaches l=2h+1
  const float* p = emb + (size_t)b * (L_ * W_ * D_)
                       + (size_t)(2 * h) * (W_ * D_)
                       + (size_t)ws * D_ + (size_t)(d0 + n);
  const int* sp = seg + b * W_ + ws;

  v2f a = {0.25f, 0.25f};                       // constant A matrix (all 0.25)
  const v8f zero8 = {0.f, 0.f, 0.f, 0.f, 0.f, 0.f, 0.f, 0.f};
  v8f c = zero8;                                // current run accumulator

  int  cur = sp[0] & (W_ - 1);
  bool firstFlush = true;                       // chunk-edge run -> atomic
  float sent_acc = 0.f;

  float* wrow = out_word + (size_t)(b * D_ + d0 + n) * W_;

  for (int t = 0; t < WCHUNK; ++t) {
    int sv = sp[t] & (W_ - 1);                  // wave-uniform
    if (sv != cur) {
      float v = c[0];                           // lane n holds value for d0+n
      if (lane < 16) {
        if (firstFlush) atomicAdd(wrow + cur, v);   // may straddle prev chunk
        else            wrow[cur] = v;              // exclusive interior run
        sent_acc += v;
      }
      firstFlush = false;
      c = zero8;
      cur = sv;
    }
    v2f bm = {p[0], p[W_ * D_]};                // this lane's two layers
    c = __builtin_amdgcn_wmma_f32_16x16x4_f32(
            /*neg_a=*/false, a, /*neg_b=*/false, bm,
            /*c_mod=*/(short)0, c, /*reuse_a=*/false, /*reuse_b=*/false);
    p += D_;
  }

  // final run of the chunk may straddle into the next chunk -> atomic
  {
    float v = c[0];
    if (lane < 16) {
      atomicAdd(wrow + cur, v);
      sent_acc += v;
      // run values already carry the 1/L factor; sent needs 1/(W*L)
      atomicAdd(out_sent + b * D_ + d0 + n, sent_acc * (1.0f / W_));
    }
  }
}

// ---------------------------------------------------------------------------
extern "C" void kernel_launch(void* const* d_in, const int* in_sizes, int n_in,
                              void* d_out, int out_size, void* d_ws, size_t ws_size,
                              hipStream_t stream) {
  const float* emb = (const float*)d_in[0];   // [B, L, W, D] f32
  const int*   seg = (const int*)d_in[1];     // [B, W] i32
  float* out = (float*)d_out;                 // word [B,D,W] then sent [B,D]

  // out_size = B*D*W + B*D = 12,607,488 (divisible by 4)
  zero_f4<<<1024, 256, 0, stream>>>((float4*)out, out_size / 4);

  dim3 grid(D_ / 128, WCHUNKS, B_);           // 6 x 4 x 32 = 768 blocks (6144 waves)
  seg_agg<<<grid, 256, 0, stream>>>(emb, seg, out, out + (size_t)B_ * D_ * W_);
}